// ANNExtractor_3925600108923
// MI455X (gfx1250) — compile-verified
//
#include <hip/hip_runtime.h>

typedef float v2f __attribute__((ext_vector_type(2)));
typedef float v8f __attribute__((ext_vector_type(8)));

namespace {
constexpr int Bn = 4;
constexpr int Tt = 256;
constexpr int FRQ = 40;
constexpr int WID = 40;          // channel count after conv1
constexpr int OC3 = 64;
constexpr int NPB = Tt * FRQ;    // 10240 pixels per batch
constexpr int NPIX = Bn * NPB;   // 40960 pixels total
constexpr int HP = Tt / 2;       // 128
constexpr int WPOOL = FRQ / 2;   // 20
constexpr int NPOOL = HP * WPOOL;       // 2560 pooled pixels per batch
constexpr int NPOOLTOT = Bn * NPOOL;    // 10240
constexpr int OUTELEMS = Bn * (Tt / 4) * (OC3 * (FRQ / 4)); // 163840
constexpr int MPAD = 48;          // 40 rows padded to 3 WMMA M-tiles
}

// -------- Kernel 1: conv1(2->40)+relu fused with Q/K/V 40x40 projections ----
// One wave per 16-pixel N-tile; f32 WMMA 16x16x4, K=40 in 10 steps.
// Weights are staged once per workgroup into LDS, zero-padded to 48 rows, so
// A-fragment reads are uniform ds_load_b64 (no divergent exec-masked loads).
__global__ __launch_bounds__(256) void k1_proj(
    const float* __restrict__ feat, const float* __restrict__ w1,
    const float* __restrict__ wkey, const float* __restrict__ wqry,
    const float* __restrict__ wval, float* __restrict__ qo,
    float* __restrict__ ko, float* __restrict__ vo) {
  __shared__ float lwq[MPAD * WID];
  __shared__ float lwk[MPAD * WID];
  __shared__ float lwv[MPAD * WID];

  for (int i = threadIdx.x; i < MPAD * WID; i += 256) {
    const bool real = i < WID * WID;
    lwq[i] = real ? wqry[i] : 0.f;
    lwk[i] = real ? wkey[i] : 0.f;
    lwv[i] = real ? wval[i] : 0.f;
  }
  __syncthreads();

  const int lane = threadIdx.x & 31;
  const int wvid = threadIdx.x >> 5;
  const int tile = blockIdx.x * 8 + wvid;
  const int col = lane & 15;
  const int hi = lane >> 4;     // which K half-pair this lane supplies
  const int off = hi << 1;
  const int p = tile * 16 + col;          // 16 | 10240: tiles never cross batch
  const int b = p / NPB;
  const int n = p - b * NPB;
  const int h = n / FRQ;
  const int w = n - h * FRQ;

  // feature (B,T,80) viewed as (B,T,2,40)
  const float f0 = feat[(size_t)(b * Tt + h) * 80 + w];
  const float f1 = feat[(size_t)(b * Tt + h) * 80 + 40 + w];

  // conv1 + relu for the 20 channels this lane feeds into B fragments
  float xb0[10], xb1[10];
#pragma unroll
  for (int s = 0; s < 10; ++s) {
    const int c0 = 4 * s + off;
    const float a0 = fmaf(w1[c0 * 2 + 1], f1, w1[c0 * 2 + 0] * f0);
    const float a1 = fmaf(w1[(c0 + 1) * 2 + 1], f1, w1[(c0 + 1) * 2 + 0] * f0);
    xb0[s] = fmaxf(a0, 0.f);
    xb1[s] = fmaxf(a1, 0.f);
  }

  v8f z = {};
  v8f accq[3] = {z, z, z};
  v8f acck[3] = {z, z, z};
  v8f accv[3] = {z, z, z};

#pragma unroll
  for (int s = 0; s < 10; ++s) {
    v2f bf;
    bf.x = xb0[s];
    bf.y = xb1[s];
    const int kx = 4 * s + off;           // even -> 8B-aligned pair reads
#pragma unroll
    for (int mt = 0; mt < 3; ++mt) {
      const int m = mt * 16 + col;        // rows 40..47 are zero padding in LDS
      const v2f aq = *(const v2f*)&lwq[m * WID + kx];
      const v2f ak = *(const v2f*)&lwk[m * WID + kx];
      const v2f av = *(const v2f*)&lwv[m * WID + kx];
      accq[mt] = __builtin_amdgcn_wmma_f32_16x16x4_f32(
          false, aq, false, bf, (short)0, accq[mt], false, false);
      acck[mt] = __builtin_amdgcn_wmma_f32_16x16x4_f32(
          false, ak, false, bf, (short)0, acck[mt], false, false);
      accv[mt] = __builtin_amdgcn_wmma_f32_16x16x4_f32(
          false, av, false, bf, (short)0, accv[mt], false, false);
    }
  }

  // C/D layout: VGPR r -> M = mt*16 + r + 8*hi, N = lane&15
#pragma unroll
  for (int mt = 0; mt < 3; ++mt) {
#pragma unroll
    for (int r = 0; r < 8; ++r) {
      const int m = mt * 16 + r + hi * 8;
      if (m < WID) {
        const size_t o = (size_t)(b * WID + m) * NPB + n;
        qo[o] = accq[mt][r];
        ko[o] = acck[mt][r];
        vo[o] = accv[mt][r];
      }
    }
  }
}

// -------- Kernel 2: 7x7 local attention (online softmax) + relu -------------
__global__ __launch_bounds__(256) void k2_attn(
    const float* __restrict__ q, const float* __restrict__ kk,
    const float* __restrict__ vv, const float* __restrict__ rel_h,
    const float* __restrict__ rel_w, float* __restrict__ out) {
  const int idx = blockIdx.x * 256 + threadIdx.x;   // [0, 40*NPIX)
  const int b = idx / (WID * NPB);
  const int c = (idx / NPB) % WID;
  const int n = idx % NPB;
  const int h = n / FRQ;
  const int w = n - h * FRQ;
  const float qv = q[idx];

  float m = -3.4e38f, s = 0.f, o = 0.f;
#pragma unroll
  for (int dh = 0; dh < 7; ++dh) {
    const int hh = h + dh - 3;
    const bool hin = ((unsigned)hh < (unsigned)Tt);
    const float rh = (c < 20) ? rel_h[c * 7 + dh] : 0.f;
#pragma unroll
    for (int dw = 0; dw < 7; ++dw) {
      const int ww = w + dw - 3;
      const bool in = hin && ((unsigned)ww < (unsigned)FRQ);
      const size_t a = (size_t)((b * WID + c) * Tt + hh) * FRQ + ww;
      // OOB taps: k projection of zero-pad = 0, but rel bias is still added;
      // v is exactly 0.
      const float kval = in ? kk[a] : 0.f;
      const float vval = in ? vv[a] : 0.f;
      const float rel = (c < 20) ? rh : rel_w[(c - 20) * 7 + dw];
      const float l = qv * (kval + rel);
      const float nm = fmaxf(m, l);
      const float e0 = __expf(m - nm);
      const float e1 = __expf(l - nm);
      s = s * e0 + e1;
      o = o * e0 + vval * e1;
      m = nm;
    }
  }
  const float r = o / s;
  out[idx] = fmaxf(r, 0.f);   // relu before the first maxpool
}

// -------- Kernel 3: maxpool 2x2 fused with conv3 (64x40 GEMM, f32 WMMA) -----
__global__ __launch_bounds__(256) void k3_conv3(
    const float* __restrict__ attn, const float* __restrict__ w3,
    float* __restrict__ c3) {
  __shared__ float lw3[OC3 * WID];
  for (int i = threadIdx.x; i < OC3 * WID; i += 256) lw3[i] = w3[i];
  __syncthreads();

  const int lane = threadIdx.x & 31;
  const int wvid = threadIdx.x >> 5;
  const int tile = blockIdx.x * 8 + wvid;
  const int col = lane & 15;
  const int hi = lane >> 4;
  const int off = hi << 1;
  const int p = tile * 16 + col;          // pooled pixel [0, 10240)
  const int b = p / NPOOL;
  const int rem = p - b * NPOOL;
  const int h2 = rem / WPOOL;
  const int w2 = rem - h2 * WPOOL;

  float xb0[10], xb1[10];
#pragma unroll
  for (int s = 0; s < 10; ++s) {
    const int c0 = 4 * s + off;
#pragma unroll
    for (int u = 0; u < 2; ++u) {
      const int c = c0 + u;
      const size_t base = ((size_t)(b * WID + c) * Tt + 2 * h2) * FRQ + 2 * w2;
      const float m0 = fmaxf(attn[base], attn[base + 1]);
      const float m1 = fmaxf(attn[base + FRQ], attn[base + FRQ + 1]);
      const float mv = fmaxf(m0, m1);
      if (u == 0) xb0[s] = mv; else xb1[s] = mv;
    }
  }

  v8f z = {};
  v8f acc[4] = {z, z, z, z};
#pragma unroll
  for (int s = 0; s < 10; ++s) {
    v2f bf;
    bf.x = xb0[s];
    bf.y = xb1[s];
    const int kx = 4 * s + off;
#pragma unroll
    for (int mt = 0; mt < 4; ++mt) {       // M = 64 -> 4 full tiles
      const int m = mt * 16 + col;
      const v2f a = *(const v2f*)&lw3[m * WID + kx];
      acc[mt] = __builtin_amdgcn_wmma_f32_16x16x4_f32(
          false, a, false, bf, (short)0, acc[mt], false, false);
    }
  }

#pragma unroll
  for (int mt = 0; mt < 4; ++mt) {
#pragma unroll
    for (int r = 0; r < 8; ++r) {
      const int m = mt * 16 + r + hi * 8;
      c3[(size_t)(b * OC3 + m) * NPOOL + rem] = acc[mt][r];
    }
  }
}

// -------- Kernel 4: maxpool 2x2 + relu + (b,c,h,w)->(b,h,c,w) + feat_len ----
__global__ __launch_bounds__(256) void k4_final(
    const float* __restrict__ c3, const int* __restrict__ feat_len,
    float* __restrict__ out) {
  const int idx = blockIdx.x * 256 + threadIdx.x;   // [0, 163840)
  const int b = idx / (OUTELEMS / Bn);
  const int r = idx % (OUTELEMS / Bn);
  const int h = r / (OC3 * (FRQ / 4));              // 0..63
  const int r2 = r % (OC3 * (FRQ / 4));
  const int c = r2 / (FRQ / 4);                     // 0..63
  const int w = r2 % (FRQ / 4);                     // 0..9

  const size_t base = ((size_t)(b * OC3 + c) * HP + 2 * h) * WPOOL + 2 * w;
  const float m0 = fmaxf(c3[base], c3[base + 1]);
  const float m1 = fmaxf(c3[base + WPOOL], c3[base + WPOOL + 1]);
  out[idx] = fmaxf(fmaxf(m0, m1), 0.f);

  if (idx < Bn) {
    out[OUTELEMS + idx] = (float)(feat_len[idx] / 4);
  }
}

extern "C" void kernel_launch(void* const* d_in, const int* in_sizes, int n_in,
                              void* d_out, int out_size, void* d_ws,
                              size_t ws_size, hipStream_t stream) {
  (void)in_sizes; (void)n_in; (void)out_size; (void)ws_size;
  const float* feat = (const float*)d_in[0];
  const int* feat_len = (const int*)d_in[1];
  const float* w1 = (const float*)d_in[2];
  const float* wkey = (const float*)d_in[3];
  const float* wqry = (const float*)d_in[4];
  const float* wval = (const float*)d_in[5];
  const float* rel_h = (const float*)d_in[6];
  const float* rel_w = (const float*)d_in[7];
  const float* w3 = (const float*)d_in[8];
  float* out = (float*)d_out;

  // Workspace partition (floats): q,k,v,attn: 40*40960 each; c3: 64*10240.
  float* ws = (float*)d_ws;
  const size_t S1 = (size_t)WID * NPIX;   // 1,638,400
  float* q = ws;
  float* kk = ws + S1;
  float* vv = ws + 2 * S1;
  float* at = ws + 3 * S1;
  float* c3 = ws + 4 * S1;                // + 655,360 -> ~28.8 MB total

  k1_proj<<<NPIX / 16 / 8, 256, 0, stream>>>(feat, w1, wkey, wqry, wval, q, kk, vv);
  k2_attn<<<(WID * NPIX) / 256, 256, 0, stream>>>(q, kk, vv, rel_h, rel_w, at);
  k3_conv3<<<NPOOLTOT / 16 / 8, 256, 0, stream>>>(at, w3, c3);
  k4_final<<<OUTELEMS / 256, 256, 0, stream>>>(c3, feat_len, out);
}